// Manifold_16303695856050
// MI455X (gfx1250) — compile-verified
//
#include <hip/hip_runtime.h>
#include <math.h>

// Problem constants (fixed by the reference)
#define NPTS 4096
#define DIMS 512
#define NCLS 128
#define ALPHA_ 0.0005f

typedef float v2f __attribute__((ext_vector_type(2)));
typedef float v8f __attribute__((ext_vector_type(8)));
typedef unsigned int v4u __attribute__((ext_vector_type(4)));
typedef int v4i __attribute__((ext_vector_type(4)));
typedef int v8i __attribute__((ext_vector_type(8)));

// ---------------------------------------------------------------------------
// Kernel 0: zero the scalar output accumulator (harness poisons d_out).
// ---------------------------------------------------------------------------
__global__ void mfld_init(float* out) { out[0] = 0.0f; }

// ---------------------------------------------------------------------------
// Kernel 1: sq[i] = sum_k x[i,k]^2   (needed for ||a-b||^2 = sq_i + sq_j - 2ab)
// 16 rows per 256-thread block; 16 threads cooperate per row.
// ---------------------------------------------------------------------------
__global__ void mfld_rowsq(const float* __restrict__ x, float* __restrict__ sq) {
  __shared__ float red[256];
  const int rb = blockIdx.x * 16;
  const int r = threadIdx.x >> 4, sub = threadIdx.x & 15;
  const float* row = x + (size_t)(rb + r) * DIMS;
  float s = 0.0f;
  for (int k = sub; k < DIMS; k += 16) { float v = row[k]; s += v * v; }
  red[threadIdx.x] = s;
  __syncthreads();
  if (sub == 0) {
    float t = 0.0f;
    for (int i = 0; i < 16; ++i) t += red[(r << 4) + i];
    sq[rb + r] = t;
  }
}

// ---------------------------------------------------------------------------
// Kernel 2: cross-entropy loss.  One block = 16 rows.  8 waves each own a
// 16-column slice of the 16x128 logits tile; K=512 accumulated with
// v_wmma_f32_16x16x4_f32.  Then per-row log-softmax + NLL -> atomicAdd.
// Dynamic LDS: xrows (32KB) + logits (8KB).
// ---------------------------------------------------------------------------
__global__ void mfld_loss(const float* __restrict__ x, const int* __restrict__ y,
                          const float* __restrict__ W, const float* __restrict__ bias,
                          float* __restrict__ out) {
  extern __shared__ char smem[];
  float* xr = (float*)smem;                          // 16 x 512
  float* lg = (float*)(smem + (size_t)16 * DIMS * 4); // 16 x 128
  const int rb = blockIdx.x * 16;

  // cooperative 16x512 row-tile load (float4, rows are 2KB-aligned)
  {
    const float4* src = (const float4*)(x + (size_t)rb * DIMS);
    float4* dst = (float4*)xr;
    for (int i = threadIdx.x; i < 16 * DIMS / 4; i += blockDim.x) dst[i] = src[i];
  }
  __syncthreads();

  const int lane = threadIdx.x & 31, wv = threadIdx.x >> 5;
  const int mn = lane & 15;      // A-row m / B-col n for this lane
  const int kh = lane >> 4;      // K-half select (ISA 16x4 f32 layout)
  const int col = wv * 16 + mn;  // output column 0..127

  v8f acc = {};
#pragma unroll 4
  for (int kk = 0; kk < DIMS; kk += 4) {
    // A-frag: lane = m + 16*(k>>1); reg0 -> k=2*kh, reg1 -> k=2*kh+1
    v2f a = *(const v2f*)&xr[mn * DIMS + kk + 2 * kh];
    // B-frag: B[k][n] = W[k*128 + col]; two k's are 128 floats apart
    v2f bfr;
    bfr.x = W[(size_t)(kk + 2 * kh) * NCLS + col];
    bfr.y = W[(size_t)(kk + 2 * kh + 1) * NCLS + col];
    acc = __builtin_amdgcn_wmma_f32_16x16x4_f32(false, a, false, bfr,
                                                (short)0, acc, false, false);
  }
  const float bv = bias[col];
#pragma unroll
  for (int v = 0; v < 8; ++v) {
    const int m = v + 8 * kh;     // C/D layout: lanes 0-15 -> M=v, 16-31 -> M=v+8
    lg[m * NCLS + col] = acc[v] + bv;
  }
  __syncthreads();

  if (threadIdx.x < 16) {
    const int r = (int)threadIdx.x;
    const float* L = lg + r * NCLS;
    float mx = L[0];
    for (int c = 1; c < NCLS; ++c) mx = fmaxf(mx, L[c]);
    float se = 0.0f;
    for (int c = 0; c < NCLS; ++c) se += expf(L[c] - mx);
    const int lbl = y[rb + r];
    const float lp = L[lbl] - mx - logf(se);
    atomicAdd(out, -lp * (1.0f / NPTS));
  }
}

// ---------------------------------------------------------------------------
// Kernel 3: regularizer.  One block = 16 rows, full 16x4096 distance block in
// LDS (256KB).  Row tile staged by the Tensor Data Mover; distance tiles via
// f32 WMMA; stable top-3 scan; rank-j order statistic via 44-bit
// (float-bits<<12 | index) binary search with wave32 shuffle reductions.
// Dynamic LDS: xrows 32KB + dist 256KB + nbr 192B  (< 320KB WGP LDS).
// ---------------------------------------------------------------------------
__global__ void mfld_reg(const float* __restrict__ x, const int* __restrict__ y,
                         const float* __restrict__ yo, const float* __restrict__ sq,
                         float* __restrict__ out) {
  extern __shared__ char smem[];
  float* xr = (float*)smem;                                        // 16 x 512
  float* dist = (float*)(smem + (size_t)16 * DIMS * 4);            // 16 x 4096
  int* nbr = (int*)(smem + (size_t)16 * DIMS * 4 + (size_t)16 * NPTS * 4);
  const int rb = blockIdx.x * 16;

  // --- TDM: DMA the 16x512 f32 row tile (global -> LDS offset 0) ---
  // D# group0: count=1, lds_addr=0, global_addr, type=2 ("image")
  // D# group1: data_size=4B (code 2), tensor_dim0=512, tensor_dim1=4096,
  //            tile_dim0=512, tile_dim1=16, tensor_dim0_stride=512
  // Groups 2/3 unused (2-D tensor) -> zeros; 6-arg clang-23 builtin form.
  if (threadIdx.x == 0) {
    unsigned long long ga = (unsigned long long)(const void*)(x + (size_t)rb * DIMS);
    v4u g0 = { 1u, 0u, (unsigned)(ga & 0xFFFFFFFFu),
               (unsigned)((ga >> 32) & 0x01FFFFFFu) | 0x80000000u };
    v8i g1 = { 0x00020000, 0x02000000, 0x10000000, 0x02000000, 16, DIMS, 0, 0 };
    v4i gz4 = { 0, 0, 0, 0 };
    v8i gz8 = { 0, 0, 0, 0, 0, 0, 0, 0 };
    __builtin_amdgcn_tensor_load_to_lds(g0, g1, gz4, gz4, gz8, 0);
  }
  __builtin_amdgcn_s_wait_tensorcnt(0);
  __syncthreads();

  const int lane = threadIdx.x & 31, wv = threadIdx.x >> 5;
  const int mn = lane & 15;   // A-row m and B-col n for this lane
  const int kh = lane >> 4;   // K-half / M-half select

  float sqr[8];
#pragma unroll
  for (int v = 0; v < 8; ++v) sqr[v] = sq[rb + v + 8 * kh];

  // --- distance tiles: each wave owns every 8th 16-column tile ---
  for (int ct = wv; ct < NPTS / 16; ct += 8) {
    const int cb = ct * 16;
    const float* bcol = x + (size_t)(cb + mn) * DIMS;  // column mn's row (L2-hot)
    v8f acc = {};
#pragma unroll 4
    for (int kk = 0; kk < DIMS; kk += 4) {
      v2f a = *(const v2f*)&xr[mn * DIMS + kk + 2 * kh];
      v2f b = *(const v2f*)&bcol[kk + 2 * kh];
      acc = __builtin_amdgcn_wmma_f32_16x16x4_f32(false, a, false, b,
                                                  (short)0, acc, false, false);
    }
    const float sqc = sq[cb + mn];
#pragma unroll
    for (int v = 0; v < 8; ++v) {
      const int m = v + 8 * kh;
      const float d2 = sqr[v] + sqc - 2.0f * acc[v];
      dist[m * NPTS + cb + mn] = (d2 > 0.0f) ? sqrtf(d2) : 0.0f;
    }
  }
  __syncthreads();

  // --- stable top-3 (ascending scan + strict '<' == argsort tie-break) ---
  if (threadIdx.x < 16) {
    const int r = (int)threadIdx.x;
    const float* Dr = dist + r * NPTS;
    float b0 = __builtin_inff(), b1 = __builtin_inff(), b2 = __builtin_inff();
    int i0 = 0, i1 = 0, i2 = 0;
    for (int k = 0; k < NPTS; ++k) {
      const float d = Dr[k];
      if (d < b0)      { b2 = b1; i2 = i1; b1 = b0; i1 = i0; b0 = d; i0 = k; }
      else if (d < b1) { b2 = b1; i2 = i1; b1 = d; i1 = k; }
      else if (d < b2) { b2 = d; i2 = k; }
    }
    nbr[r * 3 + 0] = i0; nbr[r * 3 + 1] = i1; nbr[r * 3 + 2] = i2;
  }
  __syncthreads();

  // --- per wave: rows 2*wv, 2*wv+1; for each neighbor j, the reference
  //     needs the rank-j order statistic of the row (sorted/unsorted mixing).
  for (int rr = wv * 2; rr < wv * 2 + 2; ++rr) {
    const float* Dr = dist + rr * NPTS;
    const int gi = rb + rr;
    const int yi = y[gi];
    for (int s = 0; s < 3; ++s) {
      const int j = nbr[rr * 3 + s];
      // binary search smallest 44-bit key with countLE >= j+1
      unsigned long long lo = 0, hiK = (1ull << 44) - 1;
      while (lo < hiK) {
        const unsigned long long mid = (lo + hiK) >> 1;
        int c = 0;
        for (int k = lane; k < NPTS; k += 32) {
          const unsigned long long key =
              ((unsigned long long)__float_as_uint(Dr[k]) << 12) | (unsigned)k;
          c += (key <= mid) ? 1 : 0;
        }
        for (int off = 16; off; off >>= 1) c += __shfl_xor(c, off);
        if (c >= j + 1) hiK = mid; else lo = mid + 1;
      }
      const float dval = __uint_as_float((unsigned)(lo >> 12));
      const float wgt = (y[j] == yi) ? expf(-dval) : 0.0f;  // wave-uniform
      if (wgt != 0.0f) {
        // dy[i][j] = ||y_out[i] - y_out[j]||, 128 dims over 32 lanes
        const float* yi_ = yo + (size_t)gi * NCLS;
        const float* yj_ = yo + (size_t)j * NCLS;
        float p = 0.0f;
#pragma unroll
        for (int c4 = 0; c4 < 4; ++c4) {
          const float df = yi_[lane * 4 + c4] - yj_[lane * 4 + c4];
          p += df * df;
        }
        for (int off = 16; off; off >>= 1) p += __shfl_xor(p, off);
        const float dyv = (p > 0.0f) ? sqrtf(p) : 0.0f;
        if (lane == 0) atomicAdd(out, ALPHA_ * wgt * dyv);
      }
    }
  }
}

// ---------------------------------------------------------------------------
extern "C" void kernel_launch(void* const* d_in, const int* in_sizes, int n_in,
                              void* d_out, int out_size, void* d_ws, size_t ws_size,
                              hipStream_t stream) {
  const float* x  = (const float*)d_in[0];  // x_batch  (4096 x 512)
  const int*   y  = (const int*)d_in[1];    // y_batch  (4096)
  const float* yo = (const float*)d_in[2];  // y_output (4096 x 128)
  const float* W  = (const float*)d_in[3];  // W        (512 x 128)
  const float* b  = (const float*)d_in[4];  // b        (128)
  float* out = (float*)d_out;
  float* sq  = (float*)d_ws;                // 4096 floats of scratch

  mfld_init<<<dim3(1), dim3(1), 0, stream>>>(out);
  mfld_rowsq<<<dim3(NPTS / 16), dim3(256), 0, stream>>>(x, sq);

  const size_t sm_loss = (size_t)16 * DIMS * 4 + (size_t)16 * NCLS * 4;       // 40KB
  mfld_loss<<<dim3(NPTS / 16), dim3(256), sm_loss, stream>>>(x, y, W, b, out);

  const size_t sm_reg = (size_t)16 * DIMS * 4 + (size_t)16 * NPTS * 4 + 192;  // ~288KB
  mfld_reg<<<dim3(NPTS / 16), dim3(256), sm_reg, stream>>>(x, y, yo, sq, out);
}